// ToroidalEmbedding_57750130262139
// MI455X (gfx1250) — compile-verified
//
#include <hip/hip_runtime.h>
#include <stdint.h>

// ---------------- problem constants (from reference) ----------------
#define NB     512                    // n_blocks
#define RANK   3
#define PPT    (NB * RANK)            // 1536 floats per vocab row
#define EDIM   (NB * (RANK + 1))      // 2048 output floats per token
#define WGS    128                    // threads per workgroup (4 waves)
#define TPW    4                      // tokens per workgroup (pipelined)
#define CHUNKS (PPT * 4 / (WGS * 16)) // 3 x b128 per thread per row

// builtin parameter types: b128 copy takes int4 pointers in AS1 (global) / AS3 (LDS)
typedef int v4i __attribute__((vector_size(4 * sizeof(int))));
typedef __attribute__((address_space(1))) v4i g_v4i;
typedef __attribute__((address_space(3))) v4i l_v4i;

// native clang vector for nontemporal b128 stores
typedef float v4f __attribute__((ext_vector_type(4)));

// ---- CDNA5 async global->LDS copy (ASYNCcnt-tracked DMA) ----
__device__ __forceinline__ void async_cp_b128(const void* gsrc, void* ldst) {
#if __has_builtin(__builtin_amdgcn_global_load_async_to_lds_b128)
  __builtin_amdgcn_global_load_async_to_lds_b128(
      (g_v4i*)gsrc, (l_v4i*)ldst, /*offset=*/0, /*cpol=*/0);
#else
  uint32_t loff = (uint32_t)(uintptr_t)(l_v4i*)ldst;  // LDS byte address
  uint64_t ga   = (uint64_t)(uintptr_t)gsrc;
  asm volatile("global_load_async_to_lds_b128 %0, %1, off"
               :: "v"(loff), "v"(ga) : "memory");
#endif
}

__device__ __forceinline__ void wait_async_le6() {
#if __has_builtin(__builtin_amdgcn_s_wait_asynccnt)
  __builtin_amdgcn_s_wait_asynccnt(6);
#else
  asm volatile("s_wait_asynccnt 0x6" ::: "memory");
#endif
}
__device__ __forceinline__ void wait_async_0() {
#if __has_builtin(__builtin_amdgcn_s_wait_asynccnt)
  __builtin_amdgcn_s_wait_asynccnt(0);
#else
  asm volatile("s_wait_asynccnt 0x0" ::: "memory");
#endif
}

__global__ __launch_bounds__(WGS)
void toroidal_embed_kernel(const int*   __restrict__ idx,
                           const float* __restrict__ rho,
                           const float* __restrict__ theta,
                           const float* __restrict__ gain,
                           float*       __restrict__ out,
                           int n_tokens)
{
  // double-buffered row stages (12 KB + 12 KB) + gain cache (4 KB) = 28 KB
  __shared__ __align__(16) float s_rho [2][PPT];
  __shared__ __align__(16) float s_th  [2][PPT];
  __shared__ __align__(16) float s_gain[NB * (RANK - 1)];

  const int tid  = threadIdx.x;
  const int base = blockIdx.x * TPW;

  // ---- stage gain table once: 1024 floats, 2 async b128 per thread ----
  {
    #pragma unroll
    for (int i = 0; i < 2; ++i) {
      const int f = (i * WGS + tid) * 4;
      async_cp_b128(gain + f, &s_gain[f]);
    }
  }

  // ---- prefetch token 0 rows into stage 0 ----
  if (base < n_tokens) {
    const int row = idx[base];
    const float* gr = rho   + (size_t)row * PPT;
    const float* gt = theta + (size_t)row * PPT;
    #pragma unroll
    for (int i = 0; i < CHUNKS; ++i) {
      const int f = (i * WGS + tid) * 4;
      async_cp_b128(gr + f, &s_rho[0][f]);
      async_cp_b128(gt + f, &s_th [0][f]);
    }
  }

  #pragma unroll
  for (int k = 0; k < TPW; ++k) {
    const int tok = base + k;
    if (tok >= n_tokens) break;                      // uniform per WG
    const bool havenext = (k + 1 < TPW) && (tok + 1 < n_tokens);

    if (havenext) {
      // prefetch token k+1 into the other stage (safe: its last readers
      // passed the barrier at the end of iteration k-1)
      const int row = idx[tok + 1];
      const float* gr = rho   + (size_t)row * PPT;
      const float* gt = theta + (size_t)row * PPT;
      const int st = (k + 1) & 1;
      #pragma unroll
      for (int i = 0; i < CHUNKS; ++i) {
        const int f = (i * WGS + tid) * 4;
        async_cp_b128(gr + f, &s_rho[st][f]);
        async_cp_b128(gt + f, &s_th [st][f]);
      }
      wait_async_le6();   // 6 newest copies may stay in flight; stage k done
    } else {
      wait_async_0();
    }
    __syncthreads();       // all waves' stage-k data (and gain) visible

    // ---- compute token k: 4 blocks per thread ----
    const int st = k & 1;
    float* op = out + (size_t)tok * EDIM;
    #pragma unroll
    for (int j = 0; j < NB / WGS; ++j) {
      const int b  = j * WGS + tid;
      const float r0 = s_rho[st][b * 3 + 0];
      const float r1 = s_rho[st][b * 3 + 1];
      const float r2 = s_rho[st][b * 3 + 2];
      const float t0 = s_th [st][b * 3 + 0];
      const float t1 = s_th [st][b * 3 + 1];
      const float t2 = s_th [st][b * 3 + 2];
      const float g0 = s_gain[b * 2 + 0];
      const float g1 = s_gain[b * 2 + 1];

      const float sn2 = __sinf(t2);                  // v_sin_f32
      const float c0  = __cosf(t0);                  // v_cos_f32
      const float c1  = __cosf(t1);
      const float c2  = __cosf(t2);

      const float sig0 = r0;
      const float sig1 = __builtin_fmaf(sig0, g0 * sn2, r1);
      const float sig2 = __builtin_fmaf(sig1, g1 * sn2, r2);

      v4f o;
      o.x = sig0 * c0;
      o.y = sig1 * c1;
      o.z = sig2 * c2;
      o.w = sig2 * sn2;
      // streaming output: non-temporal b128 store keeps L2 free for the
      // rho/theta gather working set (output is never re-read)
      __builtin_nontemporal_store(o, reinterpret_cast<v4f*>(op + (size_t)b * 4));
    }
    __syncthreads();       // protect this stage from the next prefetch
  }
}

extern "C" void kernel_launch(void* const* d_in, const int* in_sizes, int n_in,
                              void* d_out, int out_size, void* d_ws, size_t ws_size,
                              hipStream_t stream) {
  (void)n_in; (void)d_ws; (void)ws_size; (void)out_size;
  // setup_inputs order: idx (B*T int), rho (vocab*1536 f32),
  //                     theta (vocab*1536 f32), gain (512*2 f32)
  const int*   idx   = (const int*)  d_in[0];
  const float* rho   = (const float*)d_in[1];
  const float* theta = (const float*)d_in[2];
  const float* gain  = (const float*)d_in[3];
  float*       out   = (float*)      d_out;

  const int n_tokens = in_sizes[0];                 // 16384
  const int grid     = (n_tokens + TPW - 1) / TPW;  // 4096 WGs

  hipLaunchKernelGGL(toroidal_embed_kernel, dim3(grid), dim3(WGS), 0, stream,
                     idx, rho, theta, gain, out, n_tokens);
}